// MambaRegressor_14680198218182
// MI455X (gfx1250) — compile-verified
//
#include <hip/hip_runtime.h>
#include <hip/hip_bf16.h>
#include <cmath>
#include <cstdint>

// ---- model dimensions (match reference) ----
#define BATCH   4
#define SEQL    1024
#define HDIM    256
#define IDIM    512
#define NSTATE  16
#define RRANK   16
#define KCONV   4
#define NLAYER  4
#define EPSF    1e-5f

#define KT      32          // K-tile staged per block iteration
#define WPAD    36          // padded LDS row (floats): 16-bank-spread, keeps 16B align

typedef float v2f __attribute__((ext_vector_type(2)));
typedef float v8f __attribute__((ext_vector_type(8)));

typedef int v4i __attribute__((vector_size(16)));
typedef __attribute__((address_space(1))) v4i g_v4i;   // global int4
typedef __attribute__((address_space(3))) v4i l_v4i;   // LDS int4

// -------------------------------------------------------------------------
// Async global -> LDS 16-byte copy (CDNA5 GLOBAL_LOAD_ASYNC_TO_LDS_B128,
// tracked by ASYNCcnt). Builtin when declared, inline asm otherwise.
// -------------------------------------------------------------------------
__device__ __forceinline__ void async_copy16(const float* gsrc, float* ldst)
{
#if __has_builtin(__builtin_amdgcn_global_load_async_to_lds_b128)
    l_v4i* lp = (l_v4i*)(uintptr_t)(uint32_t)(uintptr_t)ldst;  // low 32b = LDS offset
    g_v4i* gp = (g_v4i*)(uintptr_t)gsrc;
    __builtin_amdgcn_global_load_async_to_lds_b128(gp, lp, 0, 0);
#else
    asm volatile("global_load_async_to_lds_b128 %0, %1, off"
                 :: "v"((uint32_t)(uintptr_t)ldst), "v"(gsrc)
                 : "memory");
#endif
}

__device__ __forceinline__ void async_wait0()
{
#if __has_builtin(__builtin_amdgcn_s_wait_asynccnt)
    __builtin_amdgcn_s_wait_asynccnt(0);
#else
    asm volatile("s_wait_asynccnt 0x0" ::: "memory");
#endif
}

// -------------------------------------------------------------------------
// Embedding gather: h[tok, :] = embed[ids[tok], :]
// -------------------------------------------------------------------------
__global__ void embed_kernel(const int* __restrict__ ids,
                             const float* __restrict__ embed,
                             float* __restrict__ h)
{
    int tok = blockIdx.x;
    int c   = threadIdx.x;
    int id  = ids[tok];
    h[(size_t)tok * HDIM + c] = embed[(size_t)id * HDIM + c];
}

// -------------------------------------------------------------------------
// RMSNorm: one wave (32 lanes) per token, H=256 -> 8 elems/lane
// -------------------------------------------------------------------------
__global__ void rmsnorm_kernel(const float* __restrict__ h,
                               const float* __restrict__ w,
                               float* __restrict__ out, int ntok)
{
    int tid  = blockIdx.x * blockDim.x + threadIdx.x;
    int tok  = tid >> 5;
    int lane = threadIdx.x & 31;
    if (tok >= ntok) return;
    const float* row = h + (size_t)tok * HDIM;
    float vals[8];
    float ss = 0.f;
#pragma unroll
    for (int j = 0; j < 8; ++j) {
        vals[j] = row[lane + 32 * j];
        ss += vals[j] * vals[j];
    }
#pragma unroll
    for (int off = 16; off >= 1; off >>= 1)
        ss += __shfl_xor(ss, off, 32);
    float scale = rsqrtf(ss * (1.0f / HDIM) + EPSF);
    float* orow = out + (size_t)tok * HDIM;
#pragma unroll
    for (int j = 0; j < 8; ++j)
        orow[lane + 32 * j] = vals[j] * scale * w[lane + 32 * j];
}

// -------------------------------------------------------------------------
// f32 WMMA GEMM:  Out[M,Nc] = X[M, lda](first Kd cols) @ W[Nc,Kd]^T
//   mode 0: plain    mode 1: += bias, softplus    mode 2: += res (residual)
//
// Block = 4 waves. Block tile = 64(M) x 32(N). Wave tile = 16(M) x 32(N),
// two f32 accumulators, A fragment reused across both N tiles.
// The W tile (shared by all 4 waves) is double-buffered in LDS and filled
// with async global->LDS b128 copies overlapping the WMMA stream.
// Requires M % 64 == 0 (true for all call sites: M = 4096).
// -------------------------------------------------------------------------
__global__ void gemm_wmma_f32(const float* __restrict__ X, int lda,
                              const float* __restrict__ W,
                              float* __restrict__ Out,
                              const float* __restrict__ bias,
                              const float* __restrict__ res,
                              int M, int Nc, int Kd, int mode)
{
    __shared__ float wbuf[2][32][WPAD];   // 2 x 32 rows x KT cols (padded)

    int lane = threadIdx.x;                       // 0..31
    int wave = threadIdx.y;                       // 0..3
    int tid  = wave * 32 + lane;                  // 0..127
    int n0   = blockIdx.x * 32;
    int m0   = (blockIdx.y * 4 + wave) * 16;

    int rc   = lane & 15;
    int koff = (lane >> 4) << 1;                  // 0 or 2

    const float* xrow = X + (size_t)(m0 + rc) * lda;

    // stage W[n0 .. n0+31, k0 .. k0+KT) into wbuf[buf]; 256 b128 copies / 128 thr
    auto stage = [&](int buf, int k0) {
#pragma unroll
        for (int rep = 0; rep < 2; ++rep) {
            int idx  = tid + rep * 128;           // 0..255
            int row  = idx >> 3;                  // 0..31
            int col4 = (idx & 7) << 2;            // 0,4,...,28
            if (k0 + col4 < Kd) {
                int wr = n0 + row;                // clamp for N-tail (x_proj N=48)
                if (wr >= Nc) wr = Nc - 1;
                async_copy16(W + (size_t)wr * Kd + k0 + col4, &wbuf[buf][row][col4]);
            }
        }
    };

    v8f acc0 = {0.f, 0.f, 0.f, 0.f, 0.f, 0.f, 0.f, 0.f};
    v8f acc1 = acc0;

    int nTiles = (Kd + KT - 1) / KT;

    stage(0, 0);
    async_wait0();
    __syncthreads();

    for (int t = 0; t < nTiles; ++t) {
        int k0  = t * KT;
        int cur = t & 1;
        if (t + 1 < nTiles)
            stage(cur ^ 1, k0 + KT);              // prefetch next tile (async)

        int klim = Kd - k0; if (klim > KT) klim = KT;
#pragma unroll 8
        for (int j = 0; j < klim; j += 4) {
            v2f a  = *(const v2f*)(xrow + k0 + j + koff);
            v2f b0 = *(const v2f*)&wbuf[cur][rc][j + koff];
            v2f b1 = *(const v2f*)&wbuf[cur][16 + rc][j + koff];
            acc0 = __builtin_amdgcn_wmma_f32_16x16x4_f32(
                false, a, false, b0, (short)0, acc0, false, false);
            acc1 = __builtin_amdgcn_wmma_f32_16x16x4_f32(
                false, a, false, b1, (short)0, acc1, false, false);
        }
        async_wait0();                            // next buffer fully resident
        __syncthreads();
    }

    // epilogue: D layout -> VGPR v: M = m0 + v + (lane>=16 ? 8:0), N = n0 + (lane&15)
    int mbase = m0 + ((lane >> 4) << 3);
    int nc0   = n0 + rc;
    int nc1   = n0 + 16 + rc;
#pragma unroll
    for (int v = 0; v < 8; ++v) {
        int m = mbase + v;
        float v0 = acc0[v];
        float v1 = acc1[v];
        if (mode == 1) {
            v0 += bias[nc0];
            v0 = (v0 > 20.f) ? v0 : log1pf(__expf(v0));        // softplus
            if (nc1 < Nc) {
                v1 += bias[nc1];
                v1 = (v1 > 20.f) ? v1 : log1pf(__expf(v1));
            }
        } else if (mode == 2) {
            v0 += res[(size_t)m * Nc + nc0];                   // residual add
            if (nc1 < Nc) v1 += res[(size_t)m * Nc + nc1];
        }
        Out[(size_t)m * Nc + nc0] = v0;
        if (nc1 < Nc) Out[(size_t)m * Nc + nc1] = v1;
    }
}

// -------------------------------------------------------------------------
// Causal depthwise conv (K=4) over sequence + bias + SiLU.
// -------------------------------------------------------------------------
__global__ void conv_silu_kernel(const float* __restrict__ proj,
                                 const float* __restrict__ cw,   // [I, 4]
                                 const float* __restrict__ cb,   // [I]
                                 float* __restrict__ xs)          // [B*L, I]
{
    int d  = blockIdx.x * blockDim.x + threadIdx.x;
    int bt = blockIdx.y;
    if (d >= IDIM) return;
    int b = bt / SEQL;
    int t = bt - b * SEQL;

    float acc = cb[d];
#pragma unroll
    for (int k = 0; k < KCONV; ++k) {
        int tt = t - (KCONV - 1) + k;
        if (tt >= 0)
            acc += proj[((size_t)(b * SEQL + tt)) * (2 * IDIM) + d] * cw[d * KCONV + k];
    }
    float s = acc / (1.f + __expf(-acc));
    xs[(size_t)bt * IDIM + d] = s;
}

// -------------------------------------------------------------------------
// Selective scan: one lane per (b, d) chain; N=16 states in registers.
// Fuses   y = (scan_y + xs*D) * silu(gate)   into the epilogue.
// -------------------------------------------------------------------------
__global__ void scan_kernel(const float* __restrict__ dt,    // [B*L, I]
                            const float* __restrict__ xs,    // [B*L, I]
                            const float* __restrict__ ssm,   // [B*L, R+2N]
                            const float* __restrict__ proj,  // [B*L, 2I] (gate at I+d)
                            const float* __restrict__ A_log, // [I, N]
                            const float* __restrict__ Dw,    // [I]
                            float* __restrict__ y)           // [B*L, I]
{
    int g = blockIdx.x * blockDim.x + threadIdx.x;    // 0 .. B*I-1
    int b = g / IDIM;
    int d = g - b * IDIM;

    float A[NSTATE], st[NSTATE];
#pragma unroll
    for (int n = 0; n < NSTATE; ++n) {
        A[n]  = -__expf(A_log[(size_t)d * NSTATE + n]);
        st[n] = 0.f;
    }
    float Dv = Dw[d];

    for (int t = 0; t < SEQL; ++t) {
        size_t row = (size_t)(b * SEQL + t);
        float dtv = dt[row * IDIM + d];
        float xv  = xs[row * IDIM + d];
        float gv  = proj[row * (2 * IDIM) + IDIM + d];
        float dtx = dtv * xv;
        const float* bc = ssm + row * (RRANK + 2 * NSTATE) + RRANK;  // B then C
        float acc = 0.f;
#pragma unroll
        for (int n = 0; n < NSTATE; ++n) {
            st[n] = __expf(dtv * A[n]) * st[n] + dtx * bc[n];
            acc += st[n] * bc[NSTATE + n];
        }
        float yv = (acc + xv * Dv) * (gv / (1.f + __expf(-gv)));
        y[row * IDIM + d] = yv;
    }
}

// -------------------------------------------------------------------------
// Head: final RMSNorm (pooled rows only), GELU(exact) MLP, scalar out.
// -------------------------------------------------------------------------
__global__ void head_kernel(const float* __restrict__ h,
                            const int* __restrict__ lengths,
                            const float* __restrict__ fw,
                            const float* __restrict__ w1,   // [64, H]
                            const float* __restrict__ b1,   // [64]
                            const float* __restrict__ w2,   // [64]
                            const float* __restrict__ b2,   // [1]
                            float* __restrict__ out)        // [B]
{
    __shared__ float red[HDIM];
    __shared__ float sh[HDIM];
    __shared__ float hd[HDIM / 4];
    int tid = threadIdx.x;

    for (int b = 0; b < BATCH; ++b) {
        int idx = lengths[b] - 1;
        float v = h[((size_t)b * SEQL + idx) * HDIM + tid];
        red[tid] = v * v;
        __syncthreads();
        for (int s = HDIM / 2; s >= 1; s >>= 1) {
            if (tid < s) red[tid] += red[tid + s];
            __syncthreads();
        }
        float scale = rsqrtf(red[0] * (1.0f / HDIM) + EPSF);
        __syncthreads();
        sh[tid] = v * scale * fw[tid];
        __syncthreads();
        if (tid < HDIM / 4) {
            float a = b1[tid];
            for (int i = 0; i < HDIM; ++i)
                a += sh[i] * w1[(size_t)tid * HDIM + i];
            hd[tid] = 0.5f * a * (1.f + erff(a * 0.70710678118654752f));
        }
        __syncthreads();
        if (tid == 0) {
            float a = b2[0];
            for (int j = 0; j < HDIM / 4; ++j)
                a += hd[j] * w2[j];
            out[b] = a;
        }
        __syncthreads();
    }
}

// -------------------------------------------------------------------------
// Host-side orchestration
// -------------------------------------------------------------------------
extern "C" void kernel_launch(void* const* d_in, const int* in_sizes, int n_in,
                              void* d_out, int out_size, void* d_ws, size_t ws_size,
                              hipStream_t stream)
{
    const int*   input_ids  = (const int*)  d_in[0];
    const int*   lengths    = (const int*)  d_in[1];
    const float* embed      = (const float*)d_in[2];
    const float* norm_w     = (const float*)d_in[3];
    const float* in_proj_w  = (const float*)d_in[4];
    const float* conv_w     = (const float*)d_in[5];
    const float* conv_b     = (const float*)d_in[6];
    const float* x_proj_w   = (const float*)d_in[7];
    const float* dt_proj_w  = (const float*)d_in[8];
    const float* dt_proj_b  = (const float*)d_in[9];
    const float* A_log      = (const float*)d_in[10];
    const float* Dw         = (const float*)d_in[11];
    const float* out_proj_w = (const float*)d_in[12];
    const float* final_w    = (const float*)d_in[13];
    const float* head_w1    = (const float*)d_in[14];
    const float* head_b1    = (const float*)d_in[15];
    const float* head_w2    = (const float*)d_in[16];
    const float* head_b2    = (const float*)d_in[17];
    float* out = (float*)d_out;

    const int ntok = BATCH * SEQL;               // 4096

    // workspace layout (floats)
    float* h    = (float*)d_ws;                  // [ntok, H]
    float* xn   = h    + (size_t)ntok * HDIM;    // [ntok, H]
    float* proj = xn   + (size_t)ntok * HDIM;    // [ntok, 2I]
    float* xs   = proj + (size_t)ntok * 2 * IDIM;// [ntok, I]
    float* ssm  = xs   + (size_t)ntok * IDIM;    // [ntok, R+2N]
    float* dt   = ssm  + (size_t)ntok * (RRANK + 2 * NSTATE); // [ntok, I]
    float* yb   = dt   + (size_t)ntok * IDIM;    // [ntok, I]

    dim3 gblk(32, 4);                            // 4 waves / block

    // 1) embedding gather
    embed_kernel<<<ntok, HDIM, 0, stream>>>(input_ids, embed, h);

    for (int l = 0; l < NLAYER; ++l) {
        const float* nw  = norm_w     + (size_t)l * HDIM;
        const float* ipw = in_proj_w  + (size_t)l * 2 * IDIM * HDIM;
        const float* cw  = conv_w     + (size_t)l * IDIM * KCONV;
        const float* cb  = conv_b     + (size_t)l * IDIM;
        const float* xpw = x_proj_w   + (size_t)l * (RRANK + 2 * NSTATE) * IDIM;
        const float* dpw = dt_proj_w  + (size_t)l * IDIM * RRANK;
        const float* dpb = dt_proj_b  + (size_t)l * IDIM;
        const float* al  = A_log      + (size_t)l * IDIM * NSTATE;
        const float* dv  = Dw         + (size_t)l * IDIM;
        const float* opw = out_proj_w + (size_t)l * HDIM * IDIM;

        // 2) x = rmsnorm(h)
        rmsnorm_kernel<<<(ntok * 32) / 256, 256, 0, stream>>>(h, nw, xn, ntok);

        // 3) proj = x @ in_proj_w^T           [4096 x 1024], K=256
        gemm_wmma_f32<<<dim3((2 * IDIM + 31) / 32, ntok / 64), gblk, 0, stream>>>(
            xn, HDIM, ipw, proj, nullptr, nullptr, ntok, 2 * IDIM, HDIM, 0);

        // 4) xs = silu(causal_conv(proj[:, :I]) + conv_b)
        conv_silu_kernel<<<dim3(IDIM / 256, ntok), 256, 0, stream>>>(proj, cw, cb, xs);

        // 5) ssm = xs @ x_proj_w^T            [4096 x 48], K=512
        gemm_wmma_f32<<<dim3((RRANK + 2 * NSTATE + 31) / 32, ntok / 64), gblk, 0, stream>>>(
            xs, IDIM, xpw, ssm, nullptr, nullptr, ntok, RRANK + 2 * NSTATE, IDIM, 0);

        // 6) dt = softplus(ssm[:, :R] @ dt_proj_w^T + dt_proj_b)   [4096 x 512], K=16
        gemm_wmma_f32<<<dim3((IDIM + 31) / 32, ntok / 64), gblk, 0, stream>>>(
            ssm, RRANK + 2 * NSTATE, dpw, dt, dpb, nullptr, ntok, IDIM, RRANK, 1);

        // 7) selective scan + gate/D fusion -> yb
        scan_kernel<<<(BATCH * IDIM) / 256, 256, 0, stream>>>(
            dt, xs, ssm, proj, al, dv, yb);

        // 8) h = h + yb @ out_proj_w^T        [4096 x 256], K=512 (in-place residual)
        gemm_wmma_f32<<<dim3((HDIM + 31) / 32, ntok / 64), gblk, 0, stream>>>(
            yb, IDIM, opw, h, nullptr, h, ntok, HDIM, IDIM, 2);
    }

    // 9) final norm (pooled rows only) + GELU MLP head
    head_kernel<<<1, HDIM, 0, stream>>>(h, lengths, final_w, head_w1, head_b1,
                                        head_w2, head_b2, out);
}